// WL1_81458349735985
// MI455X (gfx1250) — compile-verified
//
#include <hip/hip_runtime.h>
#include <cstdint>

// ---------------- problem constants (B=16, C=3, H=W=512) ----------------
#define Bn   16
#define Hn   512
#define Wn   512
#define HWn  (Hn * Wn)            // 262144 pixels / image
#define TOTn (Bn * HWn)           // 4194304 pixels total
#define NB1  (TOTn / 1024)        // 4096 blocks, 4 px / thread
#define BLKS_PER_IMG (HWn / 1024) // 256
// pass-3 tiling: 64 x 16 pixel tiles, 4 px / thread
#define TILX 64
#define TILY 16
#define NTX  (Wn / TILX)          // 8
#define NTY  (Hn / TILY)          // 32
#define NB3  (Bn * NTX * NTY)     // 4096
#define TSTR 72                   // LDS row stride (floats); interior at col 4 (16B aligned)

typedef float v2f __attribute__((ext_vector_type(2)));
typedef float v8f __attribute__((ext_vector_type(8)));

// ---- wave32 sum via V_WMMA_F32_16X16X4_F32 -----------------------------
// A (16x4 f32): lanes 0-15 vgpr0->K0, vgpr1->K1; lanes 16-31 -> K2,K3.
// Both A slots carry payload (2 partial sums per lane). B = ones(4x16).
// Lane-local sum of the 8 acc VGPRs gives S_low / S_high per half-wave;
// one xor-16 shuffle-add yields the full 32-lane (x2 slots) sum everywhere.
__device__ __forceinline__ float wave_sum32x2(float x0, float x1) {
    v2f a; a[0] = x0;   a[1] = x1;
    v2f b; b[0] = 1.0f; b[1] = 1.0f;
    v8f c = {};
    c = __builtin_amdgcn_wmma_f32_16x16x4_f32(
            /*neg_a=*/false, a, /*neg_b=*/false, b,
            /*c_mod=*/(short)0, c, /*reuse_a=*/false, /*reuse_b=*/false);
    float s = c[0] + c[1] + c[2] + c[3] + c[4] + c[5] + c[6] + c[7];
    s += __shfl_xor(s, 16, 32);
    return s;
}

// ---- Pass 1: residuals + mask + per-block (sum r, sum r^2), 4 px/thread
__global__ void __launch_bounds__(256)
pass1_kernel(const float* __restrict__ sr, const float* __restrict__ srema,
             const float* __restrict__ hr, float* __restrict__ rarr,
             float* __restrict__ carr, float2* __restrict__ p1) {
    const int tid = threadIdx.x;
    const int p4  = (blockIdx.x * 256 + tid) * 4;  // first of 4 consecutive px
    const int b   = p4 / HWn;                      // block never straddles images
    const int q   = p4 - b * HWn;
    const int base = b * 3 * HWn + q;              // channel stride = HWn

    float r[4] = {0, 0, 0, 0}, re[4] = {0, 0, 0, 0};
#pragma unroll
    for (int ch = 0; ch < 3; ++ch) {
        const float4 h = *(const float4*)(hr    + base + ch * HWn);
        const float4 s = *(const float4*)(sr    + base + ch * HWn);
        const float4 e = *(const float4*)(srema + base + ch * HWn);
        r[0]  += fabsf(h.x - s.x); r[1]  += fabsf(h.y - s.y);
        r[2]  += fabsf(h.z - s.z); r[3]  += fabsf(h.w - s.w);
        re[0] += fabsf(h.x - e.x); re[1] += fabsf(h.y - e.y);
        re[2] += fabsf(h.z - e.z); re[3] += fabsf(h.w - e.w);
    }
    float4 rv, cv;
    float* rp = (float*)&rv; float* cp = (float*)&cv;
#pragma unroll
    for (int i = 0; i < 4; ++i) {
        r[i]  *= (1.0f / 255.0f);
        re[i] *= (1.0f / 255.0f);
        rp[i] = r[i];
        cp[i] = (r[i] < re[i]) ? 0.0f : 255.0f * r[i];
    }
    *(float4*)(rarr + p4) = rv;
    *(float4*)(carr + p4) = cv;

    const float tr  = wave_sum32x2(r[0] + r[1], r[2] + r[3]);
    const float tr2 = wave_sum32x2(r[0] * r[0] + r[1] * r[1],
                                   r[2] * r[2] + r[3] * r[3]);

    __shared__ float w1[8], w2[8];
    const int lane = tid & 31, wv = tid >> 5;
    if (lane == 0) { w1[wv] = tr; w2[wv] = tr2; }
    __syncthreads();
    if (tid == 0) {
        float s = 0.0f, s2 = 0.0f;
#pragma unroll
        for (int i = 0; i < 8; ++i) { s += w1[i]; s2 += w2[i]; }
        p1[blockIdx.x] = make_float2(s, s2);
    }
}

// ---- Pass 2: per-image patch weight var^0.2 ----------------------------
__global__ void __launch_bounds__(256)
patchw_kernel(const float2* __restrict__ p1, float* __restrict__ pw) {
    const int b = blockIdx.x, tid = threadIdx.x;
    float s = 0.0f, s2 = 0.0f;
    for (int i = tid; i < BLKS_PER_IMG; i += 256) {
        const float2 v = p1[b * BLKS_PER_IMG + i];
        s += v.x; s2 += v.y;
    }
    __shared__ float ls[256], ls2[256];
    ls[tid] = s; ls2[tid] = s2;
    __syncthreads();
    for (int off = 128; off > 0; off >>= 1) {
        if (tid < off) { ls[tid] += ls[tid + off]; ls2[tid] += ls2[tid + off]; }
        __syncthreads();
    }
    if (tid == 0) {
        const float n   = (float)HWn;
        const float var = (ls2[0] - ls[0] * ls[0] / n) / (n - 1.0f);
        pw[b] = powf(var, 0.2f);
    }
}

// ---- Pass 3: 3x3 local variance via async-LDS tile + weighted sum ------
// 64x16 tile (4 px/thread). Halo = 66x18. Interior rows staged with
// global_load_async_to_lds_b128 (16 chunks/row, both global & LDS 16B
// aligned); the two edge columns per row use b32. Reflect folded in.
__global__ void __launch_bounds__(256)
pass3_kernel(const float* __restrict__ rarr, const float* __restrict__ carr,
             const float* __restrict__ pw, float* __restrict__ p3) {
    const int tid = threadIdx.x;
    const int bx  = blockIdx.x;
    const int tx  = bx & (NTX - 1);
    const int ty  = (bx >> 3) & (NTY - 1);
    const int b   = bx >> 8;

    __shared__ float tile[18 * TSTR];   // interior cols 4..67, halos at 3 / 68
    const float* rbase = rarr + b * HWn;
    const int gx0 = tx * TILX, gy0 = ty * TILY;

    // interior: 18 rows x 16 b128 chunks = 288 async ops
    for (int t = tid; t < 18 * 16; t += 256) {
        const int row = t >> 4, chunk = t & 15;
        int gy = gy0 - 1 + row;
        gy = (gy < 0) ? -gy : ((gy >= Hn) ? (2 * Hn - 2 - gy) : gy);   // reflect
        const float* gp = rbase + gy * Wn + gx0 + chunk * 4;           // 16B aligned
        const uint32_t lo = (uint32_t)(uintptr_t)(&tile[row * TSTR + 4 + chunk * 4]);
        asm volatile("global_load_async_to_lds_b128 %0, %1, off"
                     :: "v"(lo), "v"(gp) : "memory");
    }
    // edges: 18 rows x {left,right} = 36 async b32 ops
    for (int t = tid; t < 36; t += 256) {
        const int row = t >> 1, side = t & 1;
        int gy = gy0 - 1 + row;
        gy = (gy < 0) ? -gy : ((gy >= Hn) ? (2 * Hn - 2 - gy) : gy);
        int gx = side ? (gx0 + TILX) : (gx0 - 1);
        gx = (gx < 0) ? -gx : ((gx >= Wn) ? (2 * Wn - 2 - gx) : gx);
        const float* gp = rbase + gy * Wn + gx;
        const uint32_t lo = (uint32_t)(uintptr_t)(&tile[row * TSTR + (side ? 68 : 3)]);
        asm volatile("global_load_async_to_lds_b32 %0, %1, off"
                     :: "v"(lo), "v"(gp) : "memory");
    }
    asm volatile("s_wait_asynccnt 0" ::: "memory");   // my async ops landed
    __syncthreads();                                  // everyone's landed

    // ---- 4 outputs per thread, sliding 3x3 sums (18 LDS reads / 4 px) --
    const int y  = tid >> 4;            // 16 threads per row
    const int xq = (tid & 15) * 4;      // first of 4 x-positions
    float w[3][6];
#pragma unroll
    for (int rr = 0; rr < 3; ++rr)
#pragma unroll
        for (int cc = 0; cc < 6; ++cc)
            w[rr][cc] = tile[(y + rr) * TSTR + xq + 3 + cc];

    const float4 cv = *(const float4*)(carr + b * HWn + (gy0 + y) * Wn + gx0 + xq);
    const float* cvp = (const float*)&cv;
    const float invTot = 1.0f / (float)(Bn * 3 * HWn);
    const float pwb = pw[b] * invTot;

    float contrib[4];
#pragma unroll
    for (int i = 0; i < 4; ++i) {
        float s = 0.0f, sq = 0.0f;
#pragma unroll
        for (int rr = 0; rr < 3; ++rr) {
            const float a0 = w[rr][i], a1 = w[rr][i + 1], a2 = w[rr][i + 2];
            s  += a0 + a1 + a2;
            sq += a0 * a0 + a1 * a1 + a2 * a2;
        }
        const float lv = (sq - s * s * (1.0f / 9.0f)) * (1.0f / 8.0f);
        contrib[i] = pwb * lv * cvp[i];
    }

    const float t = wave_sum32x2(contrib[0] + contrib[1], contrib[2] + contrib[3]);
    __shared__ float wsum[8];
    const int lane = tid & 31, wv = tid >> 5;
    if (lane == 0) wsum[wv] = t;
    __syncthreads();
    if (tid == 0) {
        float ssum = 0.0f;
#pragma unroll
        for (int i = 0; i < 8; ++i) ssum += wsum[i];
        p3[bx] = ssum;
    }
}

// ---- Pass 4: deterministic final reduce of 4096 partials ---------------
__global__ void __launch_bounds__(256)
final_kernel(const float* __restrict__ p3, float* __restrict__ out) {
    const int tid = threadIdx.x;
    float s = 0.0f;
    for (int i = tid; i < NB3; i += 256) s += p3[i];
    __shared__ float ls[256];
    ls[tid] = s;
    __syncthreads();
    for (int off = 128; off > 0; off >>= 1) {
        if (tid < off) ls[tid] += ls[tid + off];
        __syncthreads();
    }
    if (tid == 0) out[0] = ls[0];
}

extern "C" void kernel_launch(void* const* d_in, const int* in_sizes, int n_in,
                              void* d_out, int out_size, void* d_ws, size_t ws_size,
                              hipStream_t stream) {
    const float* sr    = (const float*)d_in[0];
    const float* srema = (const float*)d_in[1];
    const float* hr    = (const float*)d_in[2];

    // workspace layout (floats):
    //   rarr [TOTn] | carr [TOTn] | p1 [NB1 float2] | pw [16] | p3 [NB3]
    float*  ws   = (float*)d_ws;
    float*  rarr = ws;
    float*  carr = ws + TOTn;
    float2* p1   = (float2*)(ws + 2 * TOTn);          // 8-byte aligned offset
    float*  pw   = ws + 2 * TOTn + 2 * NB1;
    float*  p3   = pw + 16;

    pass1_kernel <<<NB1, 256, 0, stream>>>(sr, srema, hr, rarr, carr, p1);
    patchw_kernel<<<Bn,  256, 0, stream>>>(p1, pw);
    pass3_kernel <<<NB3, 256, 0, stream>>>(rarr, carr, pw, p3);
    final_kernel <<<1,   256, 0, stream>>>(p3, (float*)d_out);
}